// TacrolimusPK_5961414607270
// MI455X (gfx1250) — compile-verified
//
#include <hip/hip_runtime.h>
#include <hip/hip_bf16.h>

// Batched TacrolimusPK RHS: dstate/dt for a 6-compartment model.
// Memory-bound streaming kernel: 2 rows/thread -> 16B-aligned b128 traffic,
// non-temporal hints (single-touch streams, 285MB > 192MB L2), wave32 blocks.

// clang native vectors (HIP float4/float2 structs are rejected by the
// nontemporal builtins).
typedef __attribute__((ext_vector_type(4))) float v4f;
typedef __attribute__((ext_vector_type(2))) float v2f;

struct Rhs6 { float d0, d1, d2, d3, d4, d5; };

__device__ __forceinline__ Rhs6 pk_rhs(float Ad, float Ag1, float Ag2, float Ag3,
                                       float Ac, float Ap,
                                       float ktr, float cl, float q, float vc, float vp) {
    const float k_elim = cl / vc;
    const float k12    = q  / vc;
    const float k21    = q  / vp;
    Rhs6 r;
    r.d0 = -ktr * Ad;
    r.d1 = ktr * (Ad  - Ag1);
    r.d2 = ktr * (Ag1 - Ag2);
    r.d3 = ktr * (Ag2 - Ag3);
    r.d4 = ktr * Ag3 - (k_elim + k12) * Ac + k21 * Ap;
    r.d5 = k12 * Ac - k21 * Ap;
    return r;
}

__global__ __launch_bounds__(256)
void tacrolimus_pk_rhs_kernel(const float* __restrict__ state,
                              const float* __restrict__ Ktr,
                              const float* __restrict__ CL,
                              const float* __restrict__ Q,
                              const float* __restrict__ Vc,
                              const float* __restrict__ Vp,
                              float* __restrict__ out,
                              int B) {
    const long long t  = (long long)blockIdx.x * blockDim.x + threadIdx.x;
    const long long i0 = t * 2;                  // first of 2 rows handled by this thread
    if (i0 >= B) return;

    if (i0 + 1 < B) {
        // ---- vector fast path: 2 rows = 48B of state = 3x v4f (16B aligned) ----
        const v4f* s4 = reinterpret_cast<const v4f*>(state + i0 * 6);
        const v4f a = __builtin_nontemporal_load(s4 + 0); // r0: depot,g1,g2,g3
        const v4f b = __builtin_nontemporal_load(s4 + 1); // r0: c,p | r1: depot,g1
        const v4f c = __builtin_nontemporal_load(s4 + 2); // r1: g2,g3,c,p

        // params for the 2 rows: 8B-aligned v2f loads (i0 is even)
        const v2f ktr = __builtin_nontemporal_load(reinterpret_cast<const v2f*>(Ktr + i0));
        const v2f cl  = __builtin_nontemporal_load(reinterpret_cast<const v2f*>(CL  + i0));
        const v2f q   = __builtin_nontemporal_load(reinterpret_cast<const v2f*>(Q   + i0));
        const v2f vc  = __builtin_nontemporal_load(reinterpret_cast<const v2f*>(Vc  + i0));
        const v2f vp  = __builtin_nontemporal_load(reinterpret_cast<const v2f*>(Vp  + i0));

        const Rhs6 r0 = pk_rhs(a.x, a.y, a.z, a.w, b.x, b.y,
                               ktr.x, cl.x, q.x, vc.x, vp.x);
        const Rhs6 r1 = pk_rhs(b.z, b.w, c.x, c.y, c.z, c.w,
                               ktr.y, cl.y, q.y, vc.y, vp.y);

        v4f o0, o1, o2;
        o0.x = r0.d0; o0.y = r0.d1; o0.z = r0.d2; o0.w = r0.d3;
        o1.x = r0.d4; o1.y = r0.d5; o1.z = r1.d0; o1.w = r1.d1;
        o2.x = r1.d2; o2.y = r1.d3; o2.z = r1.d4; o2.w = r1.d5;

        v4f* out4 = reinterpret_cast<v4f*>(out + i0 * 6);
        __builtin_nontemporal_store(o0, out4 + 0);
        __builtin_nontemporal_store(o1, out4 + 1);
        __builtin_nontemporal_store(o2, out4 + 2);
    } else {
        // ---- scalar tail (only if B is odd) ----
        const float* s = state + i0 * 6;
        const Rhs6 r = pk_rhs(s[0], s[1], s[2], s[3], s[4], s[5],
                              Ktr[i0], CL[i0], Q[i0], Vc[i0], Vp[i0]);
        float* o = out + i0 * 6;
        o[0] = r.d0; o[1] = r.d1; o[2] = r.d2; o[3] = r.d3; o[4] = r.d4; o[5] = r.d5;
    }
}

extern "C" void kernel_launch(void* const* d_in, const int* in_sizes, int n_in,
                              void* d_out, int out_size, void* d_ws, size_t ws_size,
                              hipStream_t stream) {
    // setup_inputs order: t, state, Ktr, CL, Q, Vc, Vp  (all float32)
    const float* state = (const float*)d_in[1];
    const float* Ktr   = (const float*)d_in[2];
    const float* CL    = (const float*)d_in[3];
    const float* Q     = (const float*)d_in[4];
    const float* Vc    = (const float*)d_in[5];
    const float* Vp    = (const float*)d_in[6];
    float* out = (float*)d_out;

    const int B = in_sizes[2];                       // batch size from Ktr length
    const long long nthreads = ((long long)B + 1) / 2; // 2 rows per thread
    const int block = 256;                           // 8 wave32 waves per block
    const long long grid = (nthreads + block - 1) / block;

    tacrolimus_pk_rhs_kernel<<<(unsigned)grid, block, 0, stream>>>(
        state, Ktr, CL, Q, Vc, Vp, out, B);
}